// VectorLagrangianKernel_7997229105665
// MI455X (gfx1250) — compile-verified
//
#include <hip/hip_runtime.h>
#include <math.h>

typedef __attribute__((ext_vector_type(2))) float v2f;
typedef __attribute__((ext_vector_type(4))) float v4f;
typedef __attribute__((ext_vector_type(8))) float v8f;

namespace {
constexpr float kDt     = 1.0f;
constexpr float kGamma  = 0.3f;
constexpr float kJitter = 1e-5f;
constexpr float kEllMin = 0.05f;
constexpr float kEllMax = 10.0f;
constexpr int   kN      = 768;
constexpr int   kT      = 24;
constexpr int   kRow2   = 2 * kN;  // 1536

struct Params {
  float sh0, sh1, w00, w01, w11, coef, pad0, pad1;
};
}  // namespace

// ---------------------------------------------------------------------------
// Setup: 96 parameter sets (t,i,j). Folds B-Sigma-B^T, Cholesky logdet, and
// the 2x2 inverse into {shift, w00,w01,w11, coef = A/sqrt(det)}.
// ---------------------------------------------------------------------------
__global__ void vl_setup(const float* __restrict__ mu,       // [24,4]
                         const float* __restrict__ Sigma,    // [24,4,4]
                         const float* __restrict__ A,        // [24,2,2]
                         const float* __restrict__ raw_ell,  // [2,2]
                         Params* __restrict__ P) {
  int idx = blockIdx.x * blockDim.x + threadIdx.x;
  if (idx >= kT * 4) return;
  int t = idx >> 2;
  int i = (idx >> 1) & 1;
  int j = idx & 1;

  float re   = raw_ell[i * 2 + j];
  float ell  = kEllMin + (kEllMax - kEllMin) / (1.0f + expf(-re));
  float ell2 = ell * ell;

  const float* Sg  = Sigma + t * 16;
  const float* mut = mu + t * 4;

  // B[i,j] rows are sparse: +dt at col 2i+r, -gamma*dt at col 2j+r.
  int a0 = 2 * i, a1 = 2 * i + 1, b0 = 2 * j, b1 = 2 * j + 1;
  float g = kGamma, g2 = kGamma * kGamma;
  float bsb00 = Sg[a0*4+a0] - g*Sg[a0*4+b0] - g*Sg[b0*4+a0] + g2*Sg[b0*4+b0];
  float bsb01 = Sg[a0*4+a1] - g*Sg[a0*4+b1] - g*Sg[b0*4+a1] + g2*Sg[b0*4+b1];
  float bsb11 = Sg[a1*4+a1] - g*Sg[a1*4+b1] - g*Sg[b1*4+a1] + g2*Sg[b1*4+b1];

  float d00 = ell2 + kJitter + 2.0f * (kDt * kDt) * bsb00;
  float d01 = 2.0f * (kDt * kDt) * bsb01;
  float d11 = ell2 + kJitter + 2.0f * (kDt * kDt) * bsb11;

  float det = d00 * d11 - d01 * d01;
  float inv = 1.0f / det;

  Params p;
  p.sh0  = kDt * (mut[2*i+0] - kGamma * mut[2*j+0]);
  p.sh1  = kDt * (mut[2*i+1] - kGamma * mut[2*j+1]);
  p.w00  = d11 * inv;
  p.w01  = -d01 * inv;
  p.w11  = d00 * inv;
  p.coef = A[t*4 + i*2 + j] / sqrtf(det);   // A * exp(-0.5*logdet)
  p.pad0 = 0.0f;
  p.pad1 = 0.0f;
  P[idx] = p;
}

__device__ __forceinline__ float kexp(float sp0, float sp1, float qx, float qy,
                                      Params p) {
  float h0 = sp0 - qx - p.sh0;
  float h1 = sp1 - qy - p.sh1;
  float m  = p.w00 * h0 * h0 + 2.0f * p.w01 * h0 * h1 + p.w11 * h1 * h1;
  return p.coef * __expf(-m);
}

// ---------------------------------------------------------------------------
// Main: one wave per (t, i, 16-row tile). Pass 1 generates values directly in
// the WMMA A-matrix 16x4 f32 layout and reduces row sums via
// V_WMMA_F32_16X16X4_F32 against a ones B matrix. Pass 2 recomputes,
// normalizes, and streams coalesced non-temporal b128 stores (output is
// write-once and > L2, so TH=NT avoids churning the 192 MB L2).
// ---------------------------------------------------------------------------
__global__ __launch_bounds__(256)
void vl_main(const float* __restrict__ S,      // [768,2]
             const Params* __restrict__ P,     // [96]
             float* __restrict__ out) {        // [24,1536,1536]
  __shared__ v4f   ldsS[kN / 2];   // (S[2e].xy, S[2e+1].xy), 6 KB
  __shared__ float ldsRN[8 * 16];  // per-wave reciprocal row sums

  const int tid  = threadIdx.x;
  const int lane = tid & 31;
  const int wave = tid >> 5;
  const int half = lane >> 4;
  const int row  = lane & 15;

  // Stage all of S into LDS once per block.
  for (int e = tid; e < kN / 2; e += 256) {
    const float* sp = S + e * 4;
    v4f v;
    v.x = sp[0]; v.y = sp[1]; v.z = sp[2]; v.w = sp[3];
    ldsS[e] = v;
  }
  __syncthreads();

  const int tile = blockIdx.x * 8 + wave;   // 2304 tiles total
  const int t    = tile / 96;
  const int rem  = tile - t * 96;
  const int i    = rem / 48;
  const int p0   = (rem - i * 48) * 16;

  const Params P0 = P[(t * 2 + i) * 2 + 0];
  const Params P1 = P[(t * 2 + i) * 2 + 1];

  // ---- Pass 1: row sums via WMMA (A layout: row = lane%16, K = 2*half+{0,1})
  v4f   sprow = ldsS[(p0 + row) >> 1];
  float sp0   = (row & 1) ? sprow.z : sprow.x;
  float sp1   = (row & 1) ? sprow.w : sprow.y;

  v8f acc = {};
  v2f ones;
  ones.x = 1.0f;
  ones.y = 1.0f;

  for (int cg = 0; cg < 192; ++cg) {            // columns 0..767 (j = 0)
    v4f sq = ldsS[cg * 2 + half];
    v2f a;
    a.x = kexp(sp0, sp1, sq.x, sq.y, P0);
    a.y = kexp(sp0, sp1, sq.z, sq.w, P0);
    acc = __builtin_amdgcn_wmma_f32_16x16x4_f32(false, a, false, ones,
                                                (short)0, acc, false, false);
  }
  for (int cg = 0; cg < 192; ++cg) {            // columns 768..1535 (j = 1)
    v4f sq = ldsS[cg * 2 + half];
    v2f a;
    a.x = kexp(sp0, sp1, sq.x, sq.y, P1);
    a.y = kexp(sp0, sp1, sq.z, sq.w, P1);
    acc = __builtin_amdgcn_wmma_f32_16x16x4_f32(false, a, false, ones,
                                                (short)0, acc, false, false);
  }

  // Every N column of D holds the row sum. Lane 0 has rows 0-7 (VGPR v = M),
  // lane 16 has rows 8-15.
  if (row == 0) {
    const int base = wave * 16 + half * 8;
#pragma unroll
    for (int v = 0; v < 8; ++v)
      ldsRN[base + v] = 1.0f / fmaxf(acc[v], 1e-8f);
  }
  __syncthreads();

  // ---- Pass 2: recompute, normalize, coalesced NT stores (16B/lane, 256B/half-wave)
  const size_t rowBase = ((size_t)t * kRow2 + (size_t)i * kN + p0) * (size_t)kRow2;
  const int    lc      = row * 4;   // column offset of this lane within 64-col slab

  for (int rp = 0; rp < 8; ++rp) {
    const int   r  = rp * 2 + half;                   // row within tile
    const float rn = ldsRN[wave * 16 + r];
    v4f   spr = ldsS[(p0 + r) >> 1];
    const float s0 = (r & 1) ? spr.z : spr.x;
    const float s1 = (r & 1) ? spr.w : spr.y;
    float* orow = out + rowBase + (size_t)r * kRow2;

    for (int c2 = 0; c2 < 12; ++c2) {                 // j = 0 half of the row
      const int q  = c2 * 64 + lc;
      v4f qa = ldsS[q >> 1];
      v4f qb = ldsS[(q >> 1) + 1];
      v4f o;
      o.x = kexp(s0, s1, qa.x, qa.y, P0) * rn;
      o.y = kexp(s0, s1, qa.z, qa.w, P0) * rn;
      o.z = kexp(s0, s1, qb.x, qb.y, P0) * rn;
      o.w = kexp(s0, s1, qb.z, qb.w, P0) * rn;
      __builtin_nontemporal_store(o, reinterpret_cast<v4f*>(orow + q));
    }
    for (int c2 = 0; c2 < 12; ++c2) {                 // j = 1 half of the row
      const int q  = c2 * 64 + lc;
      v4f qa = ldsS[q >> 1];
      v4f qb = ldsS[(q >> 1) + 1];
      v4f o;
      o.x = kexp(s0, s1, qa.x, qa.y, P1) * rn;
      o.y = kexp(s0, s1, qa.z, qa.w, P1) * rn;
      o.z = kexp(s0, s1, qb.x, qb.y, P1) * rn;
      o.w = kexp(s0, s1, qb.z, qb.w, P1) * rn;
      __builtin_nontemporal_store(o, reinterpret_cast<v4f*>(orow + kN + q));
    }
  }
}

// ---------------------------------------------------------------------------
extern "C" void kernel_launch(void* const* d_in, const int* in_sizes, int n_in,
                              void* d_out, int out_size, void* d_ws,
                              size_t ws_size, hipStream_t stream) {
  (void)in_sizes; (void)n_in; (void)out_size; (void)ws_size;
  const float* S       = (const float*)d_in[0];   // [768,2]
  const float* mu      = (const float*)d_in[1];   // [24,4]
  const float* Sigma   = (const float*)d_in[2];   // [24,4,4]
  const float* A       = (const float*)d_in[3];   // [24,2,2]
  const float* raw_ell = (const float*)d_in[4];   // [2,2]
  float*  out = (float*)d_out;
  Params* P   = (Params*)d_ws;                    // 96 * 32 B = 3 KB scratch

  vl_setup<<<1, 128, 0, stream>>>(mu, Sigma, A, raw_ell, P);
  vl_main<<<288, 256, 0, stream>>>(S, P, out);    // 2304 waves, 1 tile each
}